// Decoder_18219251269922
// MI455X (gfx1250) — compile-verified
//
#include <hip/hip_runtime.h>
#include <cstdint>

typedef float v2f __attribute__((ext_vector_type(2)));
typedef float v8f __attribute__((ext_vector_type(8)));

// ---------------------------------------------------------------------------
// Utility kernels
// ---------------------------------------------------------------------------
__global__ void zero_f32(float* __restrict__ p, int n) {
  int i = blockIdx.x * blockDim.x + threadIdx.x;
  if (i < n) p[i] = 0.0f;
}

// deg[dst] += w  (self-loop +1 folded into dinv_kernel)
__global__ void deg_kernel(const int* __restrict__ dst, const float* __restrict__ w,
                           float* __restrict__ deg, int E) {
  int e = blockIdx.x * blockDim.x + threadIdx.x;
  if (e < E) atomicAdd(&deg[dst[e]], w[e]);
}

// in-place: deg -> dinv = rsqrt(deg + 1)
__global__ void dinv_kernel(float* __restrict__ degdinv, int n) {
  int i = blockIdx.x * blockDim.x + threadIdx.x;
  if (i < n) {
    float d = degdinv[i] + 1.0f;                 // self-loop weight 1
    degdinv[i] = (d > 0.0f) ? rsqrtf(d) : 0.0f;
  }
}

// norm[e] = dinv[src] * w * dinv[dst]   (identical across all 3 layers)
__global__ void norm_kernel(const int* __restrict__ src, const int* __restrict__ dst,
                            const float* __restrict__ w, const float* __restrict__ dinv,
                            float* __restrict__ norm, int E) {
  int e = blockIdx.x * blockDim.x + threadIdx.x;
  if (e < E) norm[e] = dinv[src[e]] * w[e] * dinv[dst[e]];
}

// Zero-pad all three weight matrices into 16x16 (K x N) tiles so the WMMA
// B-fragment loads need no guards; padded columns/rows are exactly 0.
__global__ void pad_weights(const float* __restrict__ W1, const float* __restrict__ W2,
                            const float* __restrict__ W3, float* __restrict__ Wp) {
  int i = threadIdx.x;            // 0..255, one block
  int k = i >> 4, f = i & 15;
  Wp[0 * 256 + i] = (k < 2  && f < 8 ) ? W1[k * 8  + f] : 0.0f;
  Wp[1 * 256 + i] = (k < 8  && f < 16) ? W2[k * 16 + f] : 0.0f;
  Wp[2 * 256 + i] = (k < 16 && f < 3 ) ? W3[k * 3  + f] : 0.0f;
}

// ---------------------------------------------------------------------------
// Dense linear: h[N,FOUT] = x[N,FIN] @ W[FIN,FOUT] via V_WMMA_F32_16X16X4_F32.
// One wave per 16-row tile; branchless fragment loads:
//   - A: address clamped with unsigned min (garbage K-columns hit zero B rows)
//   - B: unconditional loads from the zero-padded 16x16 weight tile
//   - C: staged through per-wave LDS, then vectorized b128 coalesced store
// VGPR layouts per CDNA5 ISA 7.12.2:
//   A 16x4 : lane m (0-15) vgpr0=K0 vgpr1=K1 ; lanes 16-31 vgpr0=K2 vgpr1=K3
//   B 4x16 : vgpr0 = row K(+half*2) over lanes (N), vgpr1 = next K row
//   C 16x16: vgpr j -> M=j (lanes 0-15) / M=j+8 (lanes 16-31), N=lane&15
// ---------------------------------------------------------------------------
template <int FIN, int FOUT>
__launch_bounds__(256)
__global__ void linear_wmma(const float* __restrict__ x, const float* __restrict__ Wp,
                            float* __restrict__ h, int n) {
  __shared__ float lds[8][256];          // one 16x16 f32 tile per wave (8 waves/block)
  const int lane  = threadIdx.x & 31;
  const int wslot = threadIdx.x >> 5;
  const int wave  = (blockIdx.x * blockDim.x + threadIdx.x) >> 5;
  const int row0  = wave * 16;
  if (row0 >= n) return;                 // wave-uniform exit: EXEC all-ones while active

  const int half = lane >> 4;            // 0 = lanes 0-15, 1 = lanes 16-31
  const int mrow = lane & 15;            // A-matrix M index held by this lane
  const int ncol = lane & 15;            // B/C/D column (N) index
  const unsigned maxidx = (unsigned)n * FIN - 1u;

  v8f c = {};
#pragma unroll
  for (int kc = 0; kc < FIN; kc += 4) {
    const int ka = kc + half * 2;        // this lane holds K = ka, ka+1
    unsigned ia = (unsigned)(row0 + mrow) * FIN + (unsigned)ka;
    unsigned i0 = (ia < maxidx) ? ia : maxidx;            // branchless clamp
    unsigned i1 = (ia + 1u < maxidx) ? (ia + 1u) : maxidx;
    v2f a, b;
    a[0] = x[i0];
    a[1] = x[i1];
    b[0] = Wp[(ka + 0) * 16 + ncol];     // zero-padded: unconditional
    b[1] = Wp[(ka + 1) * 16 + ncol];
    // 8 args: (neg_a, A, neg_b, B, c_mod, C, reuse_a, reuse_b)
    c = __builtin_amdgcn_wmma_f32_16x16x4_f32(false, a, false, b, (short)0, c,
                                              false, false);
  }

  // Stage C tile into this wave's private LDS slab (no cross-wave sharing,
  // so no barrier; compiler inserts the DScnt wait).
#pragma unroll
  for (int j = 0; j < 8; ++j)
    lds[wslot][(j + half * 8) * 16 + ncol] = c[j];

  // The 16-row tile occupies a contiguous 16*FOUT span of h. Full tiles
  // (wave-uniform test) take an unguarded vectorized b128 path.
  const int base = row0 * FOUT;          // <= 1.6M, fits int
  if (row0 + 16 <= n) {
    if constexpr (FOUT % 4 == 0) {
#pragma unroll
      for (int q = 0; q < (16 * FOUT) / 128; ++q) {
        const int i4 = (lane + q * 32) * 4;
        const int r = i4 / FOUT, f = i4 % FOUT;   // f % 4 == 0 -> 16B aligned
        *(float4*)(h + base + i4) = *(const float4*)(&lds[wslot][r * 16 + f]);
      }
    } else {
      // FOUT == 3: 48 floats; lanes 0-11 each emit one gathered float4
      const int i4 = lane * 4;
      if (i4 < 16 * FOUT) {             // h buffer padded, span stays in-bounds
        float4 v;
        v.x = lds[wslot][((i4 + 0) / FOUT) * 16 + (i4 + 0) % FOUT];
        v.y = lds[wslot][((i4 + 1) / FOUT) * 16 + (i4 + 1) % FOUT];
        v.z = lds[wslot][((i4 + 2) / FOUT) * 16 + (i4 + 2) % FOUT];
        v.w = lds[wslot][((i4 + 3) / FOUT) * 16 + (i4 + 3) % FOUT];
        *(float4*)(h + base + i4) = v;
      }
    }
  } else {                               // remainder tile (never hit for N=100000)
    const int lim = n * FOUT;
    for (int i = lane; i < 16 * FOUT; i += 32)
      if (base + i < lim) h[base + i] = lds[wslot][(i / FOUT) * 16 + i % FOUT];
  }
}

// ---------------------------------------------------------------------------
// Edge scatter: acc[dst] += h[src] * norm  (one thread per edge).
// Vectorized float4 gather when rows are 16B-aligned (F % 4 == 0).
// ---------------------------------------------------------------------------
template <int F>
__launch_bounds__(256)
__global__ void scatter_kernel(const int* __restrict__ src, const int* __restrict__ dst,
                               const float* __restrict__ norm, const float* __restrict__ h,
                               float* __restrict__ acc, int E) {
  int e = blockIdx.x * blockDim.x + threadIdx.x;
  if (e >= E) return;
  const int s = src[e];
  const int d = dst[e];
  const float nr = norm[e];
  float* ad = acc + (size_t)d * F;
  if constexpr (F % 4 == 0) {
    const float4* hs4 = reinterpret_cast<const float4*>(h + (size_t)s * F);
#pragma unroll
    for (int q = 0; q < F / 4; ++q) {
      float4 v = hs4[q];
      atomicAdd(&ad[q * 4 + 0], v.x * nr);
      atomicAdd(&ad[q * 4 + 1], v.y * nr);
      atomicAdd(&ad[q * 4 + 2], v.z * nr);
      atomicAdd(&ad[q * 4 + 3], v.w * nr);
    }
  } else {
    const float* hs = h + (size_t)s * F;
#pragma unroll
    for (int f = 0; f < F; ++f) atomicAdd(&ad[f], hs[f] * nr);
  }
}

// ---------------------------------------------------------------------------
// Finalize: out = [relu](acc + h*dinv^2 + bias)   (self-loop term folded in)
// acc/out may alias (in-place), so no restrict on those.
// ---------------------------------------------------------------------------
template <int F, bool RELU>
__global__ void finalize_kernel(const float* acc, const float* __restrict__ h,
                                const float* __restrict__ dinv,
                                const float* __restrict__ bias, float* out, int n) {
  int i = blockIdx.x * blockDim.x + threadIdx.x;
  if (i >= n * F) return;
  const int node = i / F;
  const int f = i % F;
  const float di = dinv[node];
  float v = acc[i] + h[i] * (di * di) + bias[f];
  if (RELU) v = fmaxf(v, 0.0f);
  out[i] = v;
}

// ---------------------------------------------------------------------------
// Launch
// ---------------------------------------------------------------------------
extern "C" void kernel_launch(void* const* d_in, const int* in_sizes, int n_in,
                              void* d_out, int out_size, void* d_ws, size_t ws_size,
                              hipStream_t stream) {
  const float* z  = (const float*)d_in[0];   // [N,2]
  const int*   ei = (const int*)d_in[1];     // [2,E] (int32)
  const float* ew = (const float*)d_in[2];   // [E]
  const float* W1 = (const float*)d_in[3];
  const float* b1 = (const float*)d_in[4];
  const float* W2 = (const float*)d_in[5];
  const float* b2 = (const float*)d_in[6];
  const float* W3 = (const float*)d_in[7];
  const float* b3 = (const float*)d_in[8];
  float* out = (float*)d_out;

  const int N = in_sizes[0] / 2;             // 100000
  const int E = in_sizes[2];                 // 3200000
  const int* src = ei;
  const int* dst = ei + E;

  // Workspace carve-out (floats), all offsets multiples of 4 floats (16B align)
  float* ws   = (float*)d_ws;
  float* Wp   = ws;  ws += 3 * 256;          // padded 16x16 weight tiles
  float* dinv = ws;  ws += N;                // deg, then dinv in place
  float* norm = ws;  ws += E;
  float* h1   = ws;  ws += (size_t)N * 8;
  float* a1   = ws;  ws += (size_t)N * 8;    // acc1, then activations in place
  float* h2   = ws;  ws += (size_t)N * 16;
  float* a2   = ws;  ws += (size_t)N * 16;
  float* h3   = ws;  ws += (size_t)N * 4;    // stride 3, padded allocation
  float* a3   = ws;  ws += (size_t)N * 4;

  const int T = 256;                          // 8 waves per block (wave32)
  auto cdiv = [](int a, int b) { return (a + b - 1) / b; };
  const int tiles = cdiv(N, 16);
  const int lin_blocks = cdiv(tiles, T / 32);

  // --- shared normalization (one pass, reused by all 3 layers) ---
  pad_weights<<<1, 256, 0, stream>>>(W1, W2, W3, Wp);
  zero_f32<<<cdiv(N, T), T, 0, stream>>>(dinv, N);
  deg_kernel<<<cdiv(E, T), T, 0, stream>>>(dst, ew, dinv, E);
  dinv_kernel<<<cdiv(N, T), T, 0, stream>>>(dinv, N);
  norm_kernel<<<cdiv(E, T), T, 0, stream>>>(src, dst, ew, dinv, norm, E);

  // --- layer 1: 2 -> 8, relu ---
  linear_wmma<2, 8><<<lin_blocks, T, 0, stream>>>(z, Wp + 0 * 256, h1, N);
  zero_f32<<<cdiv(N * 8, T), T, 0, stream>>>(a1, N * 8);
  scatter_kernel<8><<<cdiv(E, T), T, 0, stream>>>(src, dst, norm, h1, a1, E);
  finalize_kernel<8, true><<<cdiv(N * 8, T), T, 0, stream>>>(a1, h1, dinv, b1, a1, N);

  // --- layer 2: 8 -> 16, relu ---
  linear_wmma<8, 16><<<lin_blocks, T, 0, stream>>>(a1, Wp + 1 * 256, h2, N);
  zero_f32<<<cdiv(N * 16, T), T, 0, stream>>>(a2, N * 16);
  scatter_kernel<16><<<cdiv(E, T), T, 0, stream>>>(src, dst, norm, h2, a2, E);
  finalize_kernel<16, true><<<cdiv(N * 16, T), T, 0, stream>>>(a2, h2, dinv, b2, a2, N);

  // --- layer 3: 16 -> 3, no relu ---
  linear_wmma<16, 3><<<lin_blocks, T, 0, stream>>>(a2, Wp + 2 * 256, h3, N);
  zero_f32<<<cdiv(N * 3, T), T, 0, stream>>>(a3, N * 3);
  scatter_kernel<3><<<cdiv(E, T), T, 0, stream>>>(src, dst, norm, h3, a3, E);
  finalize_kernel<3, false><<<cdiv(N * 3, T), T, 0, stream>>>(a3, h3, dinv, b3, out, N);
}